// DialogueGCN_163208757766
// MI455X (gfx1250) — compile-verified
//
#include <hip/hip_runtime.h>
#include <hip/hip_bf16.h>

// ---------------------------------------------------------------------------
// DialogueGCN on MI455X (gfx1250). All GEMMs via V_WMMA_F32_16X16X4_F32.
// L=48, D=H=G=256, A=128; only 8 live W_rel slices {0,1,2,3,96,97,98,99}.
// Factorization: agg = sum_r At_r @ (x @ W_rel[r]); nbr term collapses to
// a single broadcast vector sum_i h_i.
// ---------------------------------------------------------------------------

#define NL 48
#define DD 256
#define AA 128

typedef __attribute__((ext_vector_type(2))) float v2f;
typedef __attribute__((ext_vector_type(8))) float v8f;

__device__ __forceinline__ v8f wmma_step(v8f acc, v2f a, v2f b) {
    return __builtin_amdgcn_wmma_f32_16x16x4_f32(
        false, a, false, b, (short)0, acc, false, false);
}

// Accumulate KCNT k-steps: A row-major (&A[row][0]), B row-major (&B[0][col]).
// ISA frag layout: A: m=lane&15, k=2*(lane>>4)+j ; B: k rows 2h,2h+1 ; C row v+8h.
template<int LDB, int KCNT>
__device__ __forceinline__ v8f wmma_span(v8f acc, const float* __restrict__ ap,
                                         const float* __restrict__ bp, int half) {
    ap += 2 * half;        // &A[row][2h]
    bp += 2 * half * LDB;  // &B[2h][col]
#pragma unroll 4
    for (int kk = 0; kk < KCNT; kk += 4) {
        v2f a, b;
        a.x = ap[kk];
        a.y = ap[kk + 1];
        b.x = bp[kk * LDB];
        b.y = bp[kk * LDB + LDB];
        acc = wmma_step(acc, a, b);
    }
    return acc;
}

// C[M,N] = [A | A2] @ [B ; B2] (+bias), K split at SPLIT (SPLIT==K: no 2nd part)
template<int M, int N, int K, int SPLIT, int LDA, int LDA2, int LDB, int LDB2,
         int LDC, bool BIAS>
__device__ __forceinline__ void gemm_core(const float* __restrict__ A,
                                          const float* __restrict__ A2,
                                          const float* __restrict__ B,
                                          const float* __restrict__ B2,
                                          const float* __restrict__ bias,
                                          float* __restrict__ C) {
    const int wave   = (blockIdx.x * blockDim.x + threadIdx.x) >> 5;
    const int nwaves = (gridDim.x * blockDim.x) >> 5;
    const int lane   = threadIdx.x & 31;
    const int half   = lane >> 4;
    const int sub    = lane & 15;
    constexpr int NT = N / 16;
    constexpr int NTILES = (M / 16) * NT;

    for (int tile = wave; tile < NTILES; tile += nwaves) {
        const int mt = tile / NT, nt = tile % NT;
        const int ar = mt * 16 + sub;
        const int bc = nt * 16 + sub;
        v8f acc = {};
        acc = wmma_span<LDB, SPLIT>(acc, A + ar * LDA, B + bc, half);
        if constexpr (SPLIT < K)
            acc = wmma_span<LDB2, K - SPLIT>(acc, A2 + ar * LDA2, B2 + bc, half);
#pragma unroll
        for (int vv = 0; vv < 8; ++vv) {
            float val = acc[vv];
            if constexpr (BIAS) val += bias[bc];
            C[(mt * 16 + vv + 8 * half) * LDC + bc] = val;
        }
    }
}

// q = x@Wq (y=0), k = x@Wk (y=1); q,k contiguous in workspace
__global__ void proj_kernel(const float* __restrict__ x,
                            const float* __restrict__ Wq,
                            const float* __restrict__ Wk,
                            float* __restrict__ qk) {
    const float* B = blockIdx.y ? Wk : Wq;
    float* C = qk + blockIdx.y * (NL * AA);
    gemm_core<NL, AA, DD, DD, DD, DD, AA, AA, AA, false>(x, x, B, B, nullptr, C);
}

// Y[r] = x @ W_rel[slice(r)], r = blockIdx.y in 0..7, slice = r<4 ? r : 92+r
__global__ void rel_kernel(const float* __restrict__ x,
                           const float* __restrict__ W_rel,
                           float* __restrict__ Y) {
    const int r = blockIdx.y;
    const int slice = (r & 4) ? (92 + r) : r;  // {0,1,2,3,96,97,98,99}
    const float* B = W_rel + (long)slice * DD * DD;
    float* C = Y + (long)r * NL * DD;
    gemm_core<NL, DD, DD, DD, DD, DD, DD, DD, DD, false>(x, x, B, B, nullptr, C);
}

// h = [Aat | x] @ [Y ; W_root] + b_rgcn   (K = 384 + 256)
__global__ void h_kernel(const float* __restrict__ Aat, const float* __restrict__ x,
                         const float* __restrict__ Y, const float* __restrict__ W_root,
                         const float* __restrict__ b_rgcn, float* __restrict__ h) {
    gemm_core<NL, DD, 8 * NL + DD, 8 * NL, 8 * NL, DD, DD, DD, DD, true>(
        Aat, x, Y, W_root, b_rgcn, h);
}

// out = h @ W_self + bvec
__global__ void out_kernel(const float* __restrict__ h,
                           const float* __restrict__ W_self,
                           const float* __restrict__ bvec,
                           float* __restrict__ out) {
    gemm_core<NL, DD, DD, DD, DD, DD, DD, DD, DD, true>(h, h, W_self, W_self,
                                                        bvec, out);
}

// scores[i,t] = sum_a v[a]*tanh(q[i,a]+k[t,a]); softmax over t
__global__ void attn_kernel(const float* __restrict__ q,
                            const float* __restrict__ k,
                            const float* __restrict__ v,
                            float* __restrict__ attn) {
    const int i    = blockIdx.x;
    const int tid  = threadIdx.x;   // 128 threads = 4 waves
    const int w    = tid >> 5;
    const int lane = tid & 31;
    __shared__ float sc[NL];

    float qa[4], va[4];
#pragma unroll
    for (int c = 0; c < 4; ++c) {
        qa[c] = q[i * AA + lane + 32 * c];
        va[c] = v[lane + 32 * c];
    }
    for (int t = w; t < NL; t += 4) {     // each wave: 12 keys
        float s = 0.f;
#pragma unroll
        for (int c = 0; c < 4; ++c)
            s += va[c] * tanhf(qa[c] + k[t * AA + lane + 32 * c]);
#pragma unroll
        for (int off = 16; off > 0; off >>= 1)
            s += __shfl_xor(s, off, 32);  // wave32 reduction
        if (lane == 0) sc[t] = s;
    }
    __syncthreads();
    float m = -1e30f;
    for (int t = 0; t < NL; ++t) m = fmaxf(m, sc[t]);
    float ssum = 0.f;
    for (int t = 0; t < NL; ++t) ssum += __expf(sc[t] - m);
    if (tid < NL) attn[i * NL + tid] = __expf(sc[tid] - m) / ssum;
}

// Aat[j, r*48+i] = attn[i,j] if rel(i,j)==r else 0; r = sp_i*4 + sp_j*2 + dir
__global__ void build_kernel(const float* __restrict__ attn,
                             const int* __restrict__ speaker,
                             float* __restrict__ Aat) {
    const int idx = blockIdx.x * blockDim.x + threadIdx.x;  // 48*384 exact
    const int j  = idx / (8 * NL);
    const int kk = idx % (8 * NL);
    const int r  = kk / NL;
    const int i  = kk % NL;
    const int dir = (i < j) ? 0 : 1;
    const int rel = ((speaker[i] & 1) << 2) | ((speaker[j] & 1) << 1) | dir;
    Aat[idx] = (rel == r) ? attn[i * NL + j] : 0.0f;
}

// hsum[d] = sum_j h[j,d];  bvec[n] = b_gcn[n] + sum_d hsum[d]*W_nbr[d,n]
__global__ void sum_bvec_kernel(const float* __restrict__ h,
                                const float* __restrict__ W_nbr,
                                const float* __restrict__ b_gcn,
                                float* __restrict__ bvec) {
    __shared__ float hs[DD];
    const int t = threadIdx.x;  // 256 threads
    float s = 0.f;
    for (int j = 0; j < NL; ++j) s += h[j * DD + t];
    hs[t] = s;
    __syncthreads();
    float acc = b_gcn[t];
    for (int d = 0; d < DD; ++d) acc += hs[d] * W_nbr[d * DD + t];
    bvec[t] = acc;
}

extern "C" void kernel_launch(void* const* d_in, const int* in_sizes, int n_in,
                              void* d_out, int out_size, void* d_ws, size_t ws_size,
                              hipStream_t stream) {
    const float* x      = (const float*)d_in[0];   // [48,256]
    const int*   spk    = (const int*)  d_in[1];   // [48]
    const float* Wq     = (const float*)d_in[2];   // [256,128]
    const float* Wk     = (const float*)d_in[3];   // [256,128]
    const float* v      = (const float*)d_in[4];   // [128]
    const float* W_rel  = (const float*)d_in[5];   // [4608,256,256]
    const float* W_root = (const float*)d_in[6];   // [256,256]
    const float* b_rgcn = (const float*)d_in[7];   // [256]
    const float* W_nbr  = (const float*)d_in[8];   // [256,256]
    const float* W_self = (const float*)d_in[9];   // [256,256]
    const float* b_gcn  = (const float*)d_in[10];  // [256]
    float* out = (float*)d_out;                    // [48,256]

    float* ws   = (float*)d_ws;
    float* q    = ws;                 // 48*128
    float* k    = q    + NL * AA;     // 48*128
    float* attn = k    + NL * AA;     // 48*48
    float* Aat  = attn + NL * NL;     // 48*384
    float* Y    = Aat  + NL * 8 * NL; // 384*256
    float* h    = Y    + 8 * NL * DD; // 48*256
    float* bvec = h    + NL * DD;     // 256

    proj_kernel<<<dim3(3, 2), 256, 0, stream>>>(x, Wq, Wk, q);        // q,k
    attn_kernel<<<NL, AA, 0, stream>>>(q, k, v, attn);                // softmax
    build_kernel<<<(NL * 8 * NL) / 256, 256, 0, stream>>>(attn, spk, Aat);
    rel_kernel<<<dim3(6, 8), 256, 0, stream>>>(x, W_rel, Y);          // 8 GEMMs
    h_kernel<<<6, 256, 0, stream>>>(Aat, x, Y, W_root, b_rgcn, h);
    sum_bvec_kernel<<<1, DD, 0, stream>>>(h, W_nbr, b_gcn, bvec);
    out_kernel<<<6, 256, 0, stream>>>(h, W_self, bvec, out);
}